// GCN_2585570312241
// MI455X (gfx1250) — compile-verified
//
#include <hip/hip_runtime.h>
#include <hip/hip_bf16.h>

typedef __attribute__((ext_vector_type(2))) float v2f;
typedef __attribute__((ext_vector_type(8))) float v8f;

#define GEMM_WAVES 4   // waves per block
#define ROW_ITERS  8   // 16-row tiles processed per wave (B tile cached in VGPRs)

// ---------------------------------------------------------------- utilities
__global__ void gcn_zero_f32(float* __restrict__ p, long long n) {
  long long i = (long long)blockIdx.x * blockDim.x + threadIdx.x;
  if (i < n) p[i] = 0.0f;
}

__global__ void gcn_degrees(const int* __restrict__ src, const int* __restrict__ dst,
                            float* __restrict__ deg_out, float* __restrict__ deg_in,
                            int E) {
  int e = blockIdx.x * blockDim.x + threadIdx.x;
  if (e < E) {
    atomicAdd(&deg_out[src[e]], 1.0f);
    atomicAdd(&deg_in[dst[e]], 1.0f);
  }
}

// norm = deg > 0 ? rsqrt(deg) : 0   (deg is an integer count >= 1 when > 0)
__global__ void gcn_norm(const float* __restrict__ deg, float* __restrict__ norm, int n) {
  int i = blockIdx.x * blockDim.x + threadIdx.x;
  if (i < n) {
    float d = deg[i];
    norm[i] = (d > 0.0f) ? rsqrtf(d) : 0.0f;
  }
}

// ---------------------------------------------------------------- WMMA GEMM
// Y[r, c] = sum_k (prescale[r] * X[r, k]) * Wm[k, c]
// X: nRows x Kdim (row-major), Wm: Kdim x Mout (row-major), Y: nRows x Mout.
// Each wave owns one 16-wide column tile; it preloads the full K x 16 B-tile
// into registers (Kdim/4 x v2f) and then sweeps ROW_ITERS 16-row tiles,
// issuing Kdim/4 V_WMMA_F32_16X16X4_F32 per tile with only A-loads in flight.
template <int KDIM>
__global__ __launch_bounds__(32 * GEMM_WAVES) void gcn_gemm_wmma(
    const float* __restrict__ X, const float* __restrict__ Wm,
    const float* __restrict__ prescale, float* __restrict__ Y,
    int nRows, int Mout) {
  constexpr int KSTEPS = KDIM / 4;
  const int wave = threadIdx.x >> 5;
  const int lane = threadIdx.x & 31;
  const int nRowTiles = (nRows + 15) >> 4;
  const int tileBase = (blockIdx.x * GEMM_WAVES + wave) * ROW_ITERS;
  if (tileBase >= nRowTiles) return;  // wave-uniform: EXEC stays all-1s

  const int m  = lane & 15;   // A: row-in-tile (M). B/C/D: col-in-tile (N).
  const int kg = lane >> 4;   // K-pair selector (0 -> K={0,1}, 1 -> K={2,3})

  // ---- preload B tile into registers (shared across ROW_ITERS row tiles)
  const int col = blockIdx.y * 16 + m;
  const int colc = (col < Mout) ? col : (Mout - 1);  // clamp: cols >= Mout feed
                                                     // only unstored D columns
  v2f breg[KSTEPS];
#pragma unroll
  for (int s = 0; s < KSTEPS; ++s) {
    const int k = 4 * s + 2 * kg;
    breg[s].x = Wm[k * Mout + colc];
    breg[s].y = Wm[(k + 1) * Mout + colc];
  }

  const int tileEnd = (tileBase + ROW_ITERS < nRowTiles) ? tileBase + ROW_ITERS
                                                         : nRowTiles;
  for (int tile = tileBase; tile < tileEnd; ++tile) {  // wave-uniform loop
    int row = tile * 16 + m;
    if (row >= nRows) row = nRows - 1;  // clamped load; store guarded below
    const float scale = prescale[row];
    const float* __restrict__ xrow = X + (long long)row * KDIM + 2 * kg;

    v8f acc = {};
#pragma unroll
    for (int s = 0; s < KSTEPS; ++s) {
      // A 16x4 fp32 layout: lane m holds (M=m, K=4s+2kg) , (M=m, K=4s+2kg+1)
      v2f A = *(const v2f*)(xrow + 4 * s);
      A.x *= scale;
      A.y *= scale;
      acc = __builtin_amdgcn_wmma_f32_16x16x4_f32(
          /*neg_a=*/false, A, /*neg_b=*/false, breg[s],
          /*c_mod=*/(short)0, acc, /*reuse_a=*/false, /*reuse_b=*/false);
    }

    // D 16x16 fp32 layout: lane holds N = lane%16, VGPR r -> M = r + 8*(lane/16)
    if (col < Mout) {
#pragma unroll
      for (int r = 0; r < 8; ++r) {
        const int outRow = tile * 16 + r + 8 * kg;
        if (outRow < nRows) Y[(long long)outRow * Mout + col] = acc[r];
      }
    }
  }
}

// ---------------------------------------------------------------- scatter-add
// agg[dst[e], :] += Y[src[e], :]  -- one thread per (edge, 4-feature group).
// GROUPS is compile-time so /,% strength-reduce to mul/shift (no 64-bit div).
template <int GROUPS>
__global__ void gcn_scatter(const float* __restrict__ Y, const int* __restrict__ src,
                            const int* __restrict__ dst, float* __restrict__ agg,
                            unsigned E) {
  constexpr int COLS = GROUPS * 4;
  const unsigned idx = blockIdx.x * blockDim.x + threadIdx.x;
  if (idx >= E * GROUPS) return;
  const unsigned e = idx / GROUPS;
  const int g = (int)(idx - e * GROUPS);
  const int s = src[e];
  const int d = dst[e];
  const float4 v = *(const float4*)(Y + (long long)s * COLS + g * 4);
  float* __restrict__ out = agg + (long long)d * COLS + g * 4;
  atomicAdd(out + 0, v.x);
  atomicAdd(out + 1, v.y);
  atomicAdd(out + 2, v.z);
  atomicAdd(out + 3, v.w);
}

// ---------------------------------------------------------------- pointwise
// A[i] = A[i] * norm_in[row] + bias[col] , optional ReLU, in place
template <int COLS, bool RELU>
__global__ void gcn_pointwise(float* __restrict__ A, const float* __restrict__ norm_in,
                              const float* __restrict__ bias, unsigned n) {
  const unsigned i = blockIdx.x * blockDim.x + threadIdx.x;
  if (i >= n) return;
  const unsigned row = i / COLS;
  const unsigned c = i - row * COLS;
  float v = A[i] * norm_in[row] + bias[c];
  if (RELU) v = fmaxf(v, 0.0f);
  A[i] = v;
}

// ---------------------------------------------------------------- launch
extern "C" void kernel_launch(void* const* d_in, const int* in_sizes, int n_in,
                              void* d_out, int out_size, void* d_ws, size_t ws_size,
                              hipStream_t stream) {
  const float* feat = (const float*)d_in[0];
  const int*   src  = (const int*)d_in[1];
  const int*   dst  = (const int*)d_in[2];
  const float* W0   = (const float*)d_in[3];
  const float* b0   = (const float*)d_in[4];
  const float* W1   = (const float*)d_in[5];
  const float* b1   = (const float*)d_in[6];
  const float* W2   = (const float*)d_in[7];
  const float* b2   = (const float*)d_in[8];
  float* out = (float*)d_out;

  const int F = 128, H = 128, C = 40;
  const int N = in_sizes[0] / F;
  const int E = in_sizes[1];
  (void)in_sizes; (void)n_in; (void)out_size; (void)ws_size;

  // Workspace (floats): deg_out[N] deg_in[N] norm_out[N] norm_in[N] B0[N*H] B1[N*H]
  float* ws       = (float*)d_ws;
  float* deg_out  = ws;
  float* norm_out = ws + 2LL * N;
  float* norm_in  = ws + 3LL * N;
  float* B0       = ws + 4LL * N;
  float* B1       = B0 + (long long)N * H;

  const int TB = 256;
  auto blocks1d = [](long long n, int tb) { return (unsigned)((n + tb - 1) / tb); };

  // ---- degrees & norms (deg_out/deg_in contiguous; norm_out/norm_in contiguous)
  gcn_zero_f32<<<blocks1d(2LL * N, TB), TB, 0, stream>>>(deg_out, 2LL * N);
  gcn_degrees<<<blocks1d(E, TB), TB, 0, stream>>>(src, dst, deg_out, deg_out + N, E);
  gcn_norm<<<blocks1d(2LL * N, TB), TB, 0, stream>>>(deg_out, norm_out, 2 * N);

  const int rowTiles = (N + 15) / 16;
  const int tilesPerBlock = GEMM_WAVES * ROW_ITERS;
  const dim3 gemmBlk(32 * GEMM_WAVES);
  const dim3 gridH((rowTiles + tilesPerBlock - 1) / tilesPerBlock, H / 16);
  const dim3 gridC((rowTiles + tilesPerBlock - 1) / tilesPerBlock, (C + 15) / 16);

  // ---- layer 0: gemm(feat -> B0); agg into B1; pointwise+relu in place
  gcn_gemm_wmma<128><<<gridH, gemmBlk, 0, stream>>>(feat, W0, norm_out, B0, N, H);
  gcn_zero_f32<<<blocks1d((long long)N * H, TB), TB, 0, stream>>>(B1, (long long)N * H);
  gcn_scatter<32><<<blocks1d((long long)E * 32, TB), TB, 0, stream>>>(B0, src, dst, B1, E);
  gcn_pointwise<128, true><<<blocks1d((long long)N * H, TB), TB, 0, stream>>>(
      B1, norm_in, b0, (unsigned)(N * H));

  // ---- layer 1: gemm(B1 -> B0); B1 freed after gemm, reuse for agg
  gcn_gemm_wmma<128><<<gridH, gemmBlk, 0, stream>>>(B1, W1, norm_out, B0, N, H);
  gcn_zero_f32<<<blocks1d((long long)N * H, TB), TB, 0, stream>>>(B1, (long long)N * H);
  gcn_scatter<32><<<blocks1d((long long)E * 32, TB), TB, 0, stream>>>(B0, src, dst, B1, E);
  gcn_pointwise<128, true><<<blocks1d((long long)N * H, TB), TB, 0, stream>>>(
      B1, norm_in, b1, (unsigned)(N * H));

  // ---- layer 2 (C=40): gemm(B1 -> B0, stride C); scatter straight into d_out
  gcn_gemm_wmma<128><<<gridC, gemmBlk, 0, stream>>>(B1, W2, norm_out, B0, N, C);
  gcn_zero_f32<<<blocks1d((long long)N * C, TB), TB, 0, stream>>>(out, (long long)N * C);
  gcn_scatter<10><<<blocks1d((long long)E * 10, TB), TB, 0, stream>>>(B0, src, dst, out, E);
  gcn_pointwise<40, false><<<blocks1d((long long)N * C, TB), TB, 0, stream>>>(
      out, norm_in, b2, (unsigned)(N * C));
}